// DeltaNetDecodeTraceable_37417755083571
// MI455X (gfx1250) — compile-verified
//
#include <hip/hip_runtime.h>
#include <math.h>

typedef float v2f __attribute__((ext_vector_type(2)));
typedef float v8f __attribute__((ext_vector_type(8)));

#define H_IN 2048
#define EPSF 1e-6f

__device__ __forceinline__ float sigmoidf_(float x) { return 1.0f / (1.0f + __expf(-x)); }
__device__ __forceinline__ float siluf_(float x)    { return x * sigmoidf_(x); }

// ---------------------------------------------------------------------------
// FP32 GEMV tile via V_WMMA_F32_16X16X4_F32.
// Computes y[row0+m] = dot(W[row0+m, 0:H], x[0:H]) for m = 0..15 (one wave).
// A[m][k] = W tile, B[k][n] = x broadcast across n -> D[m][n] identical in n.
// K-slot assignment chosen so each lane feeds 2 WMMAs from one float4 load:
//   lanes 0-15  supply K0,K1 from cols h0..h0+3
//   lanes 16-31 supply K2,K3 from cols h0+4..h0+7
// Result: lane 0 holds rows row0+0..7 in acc[0..7], lane 16 rows row0+8..15.
// ---------------------------------------------------------------------------
__device__ __forceinline__ v8f gemv_tile_wmma(const float* __restrict__ W,
                                              const float* __restrict__ x,
                                              int row0, int H) {
  const int lane = threadIdx.x & 31;
  const int r    = lane & 15;
  const int hi   = lane >> 4;  // 0 or 1
  const float* wrow = W + (size_t)(row0 + r) * (size_t)H + (hi << 2);
  const float* xoff = x + (hi << 2);
  v8f acc = {0.f, 0.f, 0.f, 0.f, 0.f, 0.f, 0.f, 0.f};
  for (int h0 = 0; h0 < H; h0 += 8) {
    float4 w4 = *(const float4*)(wrow + h0);
    float4 x4 = *(const float4*)(xoff + h0);
    v2f a0 = {w4.x, w4.y};
    v2f b0 = {x4.x, x4.y};
    acc = __builtin_amdgcn_wmma_f32_16x16x4_f32(false, a0, false, b0,
                                                (short)0, acc, false, false);
    v2f a1 = {w4.z, w4.w};
    v2f b1 = {x4.z, x4.w};
    acc = __builtin_amdgcn_wmma_f32_16x16x4_f32(false, a1, false, b1,
                                                (short)0, acc, false, false);
  }
  return acc;
}

// ---------------------------------------------------------------------------
// Plain WMMA-GEMV: y = W @ x   (block = 256 threads = 8 waves, 16 rows/wave)
// ---------------------------------------------------------------------------
__global__ void k_gemv(const float* __restrict__ W, const float* __restrict__ x,
                       float* __restrict__ y, int H) {
  const int wave = threadIdx.x >> 5;
  const int row0 = blockIdx.x * 128 + wave * 16;
  v8f acc = gemv_tile_wmma(W, x, row0, H);
  const int lane = threadIdx.x & 31;
  if ((lane & 15) == 0) {
    const int base = row0 + (lane >> 4) * 8;
#pragma unroll
    for (int j = 0; j < 8; ++j) y[base + j] = acc[j];
  }
}

// ---------------------------------------------------------------------------
// mixed = W_qkv @ x, then fused depthwise conv (4 taps) + SiLU, new_conv out.
// ---------------------------------------------------------------------------
__global__ void k_qkv_conv(const float* __restrict__ W_qkv, const float* __restrict__ x,
                           const float* __restrict__ conv_state,
                           const float* __restrict__ conv_w,
                           float* __restrict__ act, float* __restrict__ new_conv) {
  const int wave = threadIdx.x >> 5;
  const int row0 = blockIdx.x * 128 + wave * 16;
  v8f acc = gemv_tile_wmma(W_qkv, x, row0, H_IN);
  const int lane = threadIdx.x & 31;
  if ((lane & 15) == 0) {
    const int base = row0 + (lane >> 4) * 8;
#pragma unroll
    for (int j = 0; j < 8; ++j) {
      const int c = base + j;
      const float m  = acc[j];
      const float c0 = conv_state[c * 3 + 0];
      const float c1 = conv_state[c * 3 + 1];
      const float c2 = conv_state[c * 3 + 2];
      const float w0 = conv_w[c * 4 + 0];
      const float w1 = conv_w[c * 4 + 1];
      const float w2 = conv_w[c * 4 + 2];
      const float w3 = conv_w[c * 4 + 3];
      const float co = c0 * w0 + c1 * w1 + c2 * w2 + m * w3;
      act[c] = siluf_(co);
      new_conv[c * 3 + 0] = c1;
      new_conv[c * 3 + 1] = c2;
      new_conv[c * 3 + 2] = m;
    }
  }
}

// ---------------------------------------------------------------------------
// Gate projections: 64 rows (32 of W_b -> beta, 32 of W_a -> g_t), wave/row.
// ---------------------------------------------------------------------------
__global__ void k_gates(const float* __restrict__ W_b, const float* __restrict__ W_a,
                        const float* __restrict__ x,
                        const float* __restrict__ A_log, const float* __restrict__ dt_bias,
                        float* __restrict__ betas, float* __restrict__ gts) {
  const int wave = (blockIdx.x * blockDim.x + threadIdx.x) >> 5;  // 0..63
  const int lane = threadIdx.x & 31;
  const float* W = (wave < 32) ? (W_b + (size_t)wave * H_IN)
                               : (W_a + (size_t)(wave - 32) * H_IN);
  float s = 0.f;
  for (int i = lane; i < H_IN; i += 32) s = fmaf(W[i], x[i], s);
#pragma unroll
  for (int m = 16; m > 0; m >>= 1) s += __shfl_xor(s, m, 32);
  if (lane == 0) {
    if (wave < 32) {
      betas[wave] = sigmoidf_(s);
    } else {
      const int h = wave - 32;
      const float t  = s + dt_bias[h];
      const float sp = (t > 20.f) ? t : log1pf(__expf(t));  // softplus
      gts[h] = __expf(-__expf(A_log[h]) * sp);              // exp(g)
    }
  }
}

__device__ __forceinline__ float block_reduce_128(float val, volatile float* red, int t) {
  __syncthreads();
  red[t] = val;
  __syncthreads();
  for (int s = 64; s > 0; s >>= 1) {
    if (t < s) red[t] = red[t] + red[t + s];
    __syncthreads();
  }
  const float r = red[0];
  __syncthreads();
  return r;
}

// ---------------------------------------------------------------------------
// Per-head recurrent update. Block = head (128 threads, thread = column v).
// ---------------------------------------------------------------------------
__global__ void k_state(const float* __restrict__ state, const float* __restrict__ act,
                        const float* __restrict__ z,
                        const float* __restrict__ betas, const float* __restrict__ gts,
                        const float* __restrict__ norm_w,
                        float* __restrict__ new_state, float* __restrict__ normed) {
  __shared__ float red[128];
  __shared__ float s_kn[128];
  __shared__ float s_qn[128];
  const int h = blockIdx.x;
  const int t = threadIdx.x;
  const int g16 = h >> 1;  // q/k head groups are repeated 2x

  const float q = act[g16 * 128 + t];
  const float k = act[2048 + g16 * 128 + t];
  const float v = act[4096 + h * 128 + t];

  const float sqq = block_reduce_128(q * q, red, t);
  const float sqk = block_reduce_128(k * k, red, t);
  const float qn = q * rsqrtf(sqq + EPSF) * 0.08838834764831845f;  // /sqrt(128)
  const float kn = k * rsqrtf(sqk + EPSF);
  s_kn[t] = kn;
  s_qn[t] = qn;
  __syncthreads();

  const float gt   = gts[h];
  const float beta = betas[h];
  const float* S = state + (size_t)h * 16384;

  // kv_mem[v] = g_t * sum_k S[k][v] * kn[k]
  float acc = 0.f;
#pragma unroll 4
  for (int kk = 0; kk < 128; ++kk) acc = fmaf(S[kk * 128 + t], s_kn[kk], acc);
  acc *= gt;
  const float delta = (v - acc) * beta;  // delta[v], thread-local

  // new_state[k][v] = g_t*S[k][v] + kn[k]*delta[v]; out[v] = sum_k ns*qn[k]
  float* NS = new_state + (size_t)h * 16384;
  float out = 0.f;
#pragma unroll 4
  for (int kk = 0; kk < 128; ++kk) {
    const float ns = fmaf(s_kn[kk], delta, S[kk * 128 + t] * gt);
    NS[kk * 128 + t] = ns;
    out = fmaf(ns, s_qn[kk], out);
  }

  const float zz    = z[h * 128 + t];
  const float gated = out * siluf_(zz);
  const float msq   = block_reduce_128(gated * gated, red, t) * (1.0f / 128.0f);
  normed[h * 128 + t] = gated * rsqrtf(msq + EPSF) * norm_w[t];
}

// ---------------------------------------------------------------------------
extern "C" void kernel_launch(void* const* d_in, const int* in_sizes, int n_in,
                              void* d_out, int out_size, void* d_ws, size_t ws_size,
                              hipStream_t stream) {
  const float* x       = (const float*)d_in[0];
  const float* state   = (const float*)d_in[1];
  const float* convst  = (const float*)d_in[2];
  const float* W_qkv   = (const float*)d_in[3];
  const float* W_z     = (const float*)d_in[4];
  const float* W_b     = (const float*)d_in[5];
  const float* W_a     = (const float*)d_in[6];
  const float* conv_w  = (const float*)d_in[7];
  const float* A_log   = (const float*)d_in[8];
  const float* dt_bias = (const float*)d_in[9];
  const float* norm_w  = (const float*)d_in[10];
  const float* W_out   = (const float*)d_in[11];

  float* out_vec   = (float*)d_out;            // 2048
  float* new_state = out_vec + 2048;           // 32*128*128
  float* new_conv  = new_state + 32 * 128 * 128;  // 8192*3

  float* ws    = (float*)d_ws;
  float* act   = ws;           // 8192  (silu(conv_out))
  float* zbuf  = act + 8192;   // 4096
  float* nrm   = zbuf + 4096;  // 4096
  float* betas = nrm + 4096;   // 32
  float* gts   = betas + 32;   // 32

  k_gates<<<8, 256, 0, stream>>>(W_b, W_a, x, A_log, dt_bias, betas, gts);
  k_qkv_conv<<<64, 256, 0, stream>>>(W_qkv, x, convst, conv_w, act, new_conv);
  k_gemv<<<32, 256, 0, stream>>>(W_z, x, zbuf, 2048);
  k_state<<<32, 128, 0, stream>>>(state, act, zbuf, betas, gts, norm_w, new_state, nrm);
  k_gemv<<<16, 256, 0, stream>>>(W_out, nrm, out_vec, 4096);
}